// Gated_spiral_dw_21878563406307
// MI455X (gfx1250) — compile-verified
//
#include <hip/hip_runtime.h>

#define N_NODES 50000
#define SEQ     12
#define CH      128
#define BS      4
#define TILE    16

typedef __attribute__((ext_vector_type(2))) float v2f;
typedef __attribute__((ext_vector_type(4))) float v4f;
typedef __attribute__((ext_vector_type(8))) float v8f;

__global__ __launch_bounds__(256) void gated_spiral_kernel(
    const float* __restrict__ x,        // [BS, N, CH]
    const int*   __restrict__ indices,  // [N, SEQ]
    const float* __restrict__ weight,   // [N, SEQ]
    const float* __restrict__ gate_w,   // [CH, CH]  (gate[n,d] = sum_c x[n,c]*gate_w[d,c])
    const float* __restrict__ gate_b,   // [CH]
    float*       __restrict__ out)      // [BS, N, CH]
{
    __shared__ float xs[TILE][CH];       // 8 KB: x tile, WMMA A operand source
    __shared__ float spiral[TILE][CH];   // 8 KB: gather-reduce, then fused result
    __shared__ float w_lds[TILE][SEQ];
    __shared__ int   idx_lds[TILE][SEQ];

    const int tid       = threadIdx.x;
    const int node_base = blockIdx.x * TILE;
    const int b         = blockIdx.y;
    const float* xb     = x + (size_t)b * N_NODES * CH;

    // ---- Phase A: stage x tile [16 x 128] into LDS (512 float4, 2 per thread) ----
    {
        const v4f* src = (const v4f*)(xb + (size_t)node_base * CH);
        v4f* dst = (v4f*)&xs[0][0];
        dst[tid]       = src[tid];
        dst[tid + 256] = src[tid + 256];
    }
    if (tid < TILE * SEQ) {
        const int n = tid / SEQ, s = tid % SEQ;
        idx_lds[n][s] = indices[(node_base + n) * SEQ + s];
        w_lds[n][s]   = weight [(node_base + n) * SEQ + s];
    }
    __syncthreads();

    // ---- Phase B: gate GEMM tile via V_WMMA_F32_16X16X4_F32 ----
    // wave w owns output tile rows = 16 nodes, cols d = [16w, 16w+16)
    const int lane  = tid & 31;
    const int wave  = tid >> 5;
    const int row   = lane & 15;   // M for A / N for B / N for C
    const int half  = lane >> 4;   // 0: K{0,1}, 1: K{2,3} within each 4-wide K step
    const int dbase = wave * 16;

    v8f acc = {};
    const float* aptr = &xs[row][2 * half];                              // 8B aligned
    const float* bptr = gate_w + (size_t)(dbase + row) * CH + 2 * half;  // 8B aligned
#pragma unroll
    for (int k = 0; k < CH / 4; ++k) {
        v2f a = *(const v2f*)(aptr + 4 * k);   // ds_load_b64 (merged by compiler)
        v2f bv = *(const v2f*)(bptr + 4 * k);  // global_load_b64 (gate_w is L2-resident)
        acc = __builtin_amdgcn_wmma_f32_16x16x4_f32(
            /*neg_a=*/false, a, /*neg_b=*/false, bv,
            /*c_mod=*/(short)0, acc, /*reuse_a=*/false, /*reuse_b=*/false);
    }

    // ---- Phase C: spiral gather-reduce -> LDS ----
    // thread t handles node n = t/16, feature slice fb..fb+7 (512B/row coalesced)
    {
        const int n  = tid >> 4;
        const int fb = (tid & 15) * 8;
        float a0 = 0.f, a1 = 0.f, a2 = 0.f, a3 = 0.f;
        float a4 = 0.f, a5 = 0.f, a6 = 0.f, a7 = 0.f;
#pragma unroll
        for (int s = 0; s < SEQ; ++s) {
            const int   j = idx_lds[n][s];
            const float w = w_lds[n][s];
            const v4f* p = (const v4f*)(xb + (size_t)j * CH + fb);
            const v4f v0 = p[0];
            const v4f v1 = p[1];
            a0 = fmaf(w, v0.x, a0); a1 = fmaf(w, v0.y, a1);
            a2 = fmaf(w, v0.z, a2); a3 = fmaf(w, v0.w, a3);
            a4 = fmaf(w, v1.x, a4); a5 = fmaf(w, v1.y, a5);
            a6 = fmaf(w, v1.z, a6); a7 = fmaf(w, v1.w, a7);
        }
        v4f* q = (v4f*)&spiral[n][fb];
        q[0] = (v4f){a0, a1, a2, a3};
        q[1] = (v4f){a4, a5, a6, a7};
    }
    __syncthreads();

    // ---- Phase D: fuse gate (WMMA C layout) into the LDS tile in place ----
    // C/D layout: VGPR j, lane l -> node m = j + 8*(l>=16), d = dbase + (l&15)
    // Each (m, d) element has exactly one owning lane -> race-free RMW.
    {
        const float gb = gate_b[dbase + row];
#pragma unroll
        for (int j = 0; j < 8; ++j) {
            const int m = j + 8 * half;
            spiral[m][dbase + row] *= (acc[j] + gb);
        }
    }
    __syncthreads();

    // ---- Phase E: coalesced non-temporal stores of the fused 8 KB tile ----
    // NT hint keeps the write-once output from evicting x (which must stay
    // L2-resident to serve the 1.2 GB of gather traffic from L2, not HBM).
    {
        v4f* dst = (v4f*)(out + ((size_t)b * N_NODES + node_base) * CH);
        const v4f* srcl = (const v4f*)&spiral[0][0];
        __builtin_nontemporal_store(srcl[tid],       &dst[tid]);
        __builtin_nontemporal_store(srcl[tid + 256], &dst[tid + 256]);
    }
}

extern "C" void kernel_launch(void* const* d_in, const int* in_sizes, int n_in,
                              void* d_out, int out_size, void* d_ws, size_t ws_size,
                              hipStream_t stream) {
    const float* x       = (const float*)d_in[0];
    const int*   indices = (const int*)  d_in[1];
    const float* weight  = (const float*)d_in[2];
    const float* gate_w  = (const float*)d_in[3];
    const float* gate_b  = (const float*)d_in[4];
    float*       out     = (float*)d_out;

    dim3 grid(N_NODES / TILE, BS);   // 3125 x 4 blocks, 256 threads (8 waves) each
    gated_spiral_kernel<<<grid, dim3(256), 0, stream>>>(
        x, indices, weight, gate_w, gate_b, out);
}